// CircleLayer_80376017977658
// MI455X (gfx1250) — compile-verified
//
#include <hip/hip_runtime.h>
#include <math.h>

typedef __attribute__((ext_vector_type(2))) float v2f;
typedef __attribute__((ext_vector_type(8))) float v8f;

#define PARTS   8
#define NNEI    128
#define TLEN    20
#define DRES    64
#define TWO_PI_F 6.28318530717958647692f

// One block per batch element. 128 threads = 4 wave32.
// Phase 1: per-neighbor features + LDS atomic bin stats.
// Phase 2: dense ReLU half of the output.
// Phase 3: per-wave V_WMMA_F32_16X16X4_F32 segmented sum of f_resonance.
__launch_bounds__(128)
__global__ void circle_layer_kernel(const float* __restrict__ ego,   // [B, T, 2]
                                    const float* __restrict__ nei,   // [B, N, T, 2]
                                    const float* __restrict__ fres,  // [B, N, D]
                                    const float* __restrict__ Wce,   // [2, D]
                                    const float* __restrict__ bce,   // [D]
                                    float* __restrict__ out)         // [B, P, 2D]
{
    __shared__ int   s_idx[NNEI];     // angle bin per neighbor, -1 = invalid
    __shared__ float s_cnt[PARTS];
    __shared__ float s_sd[PARTS];     // sum of distances per bin
    __shared__ float s_sr[PARTS];     // sum of directions per bin
    __shared__ float s_inv[PARTS];    // 1 / (count + 1e-4)

    const int b    = blockIdx.x;
    const int tid  = threadIdx.x;     // 0..127
    const int lane = tid & 31;
    const int wave = tid >> 5;        // 0..3

    if (tid < PARTS) { s_cnt[tid] = 0.f; s_sd[tid] = 0.f; s_sr[tid] = 0.f; }
    __syncthreads();

    // ---------------- Phase 1: per-neighbor features (thread == neighbor) ----
    {
        const int n = tid;
        const float4* p = (const float4*)(nei + ((size_t)b * NNEI + n) * (TLEN * 2));
        float sum = 0.f, lastx = 0.f, lasty = 0.f;
        #pragma unroll
        for (int q = 0; q < (TLEN * 2) / 4; ++q) {     // 10 x b128 loads, contiguous
            float4 v = p[q];
            sum += v.x + v.y + v.z + v.w;
            if (q == (TLEN * 2) / 4 - 1) { lastx = v.z; lasty = v.w; }
        }
        const float ex = ego[(size_t)b * TLEN * 2 + (TLEN - 1) * 2 + 0];
        const float ey = ego[(size_t)b * TLEN * 2 + (TLEN - 1) * 2 + 1];
        const float rx = lastx - ex;
        const float ry = lasty - ey;

        int idx = -1;
        if (sum != 0.0f) {                             // validity mask
            float dist = sqrtf(rx * rx + ry * ry);
            float dir  = atan2f(rx, ry);               // jnp.arctan2(x, y)
            dir -= floorf(dir * (1.0f / TWO_PI_F)) * TWO_PI_F;   // jnp.mod(. , 2pi)
            int k = (int)(dir * ((float)PARTS / TWO_PI_F));
            if (k > PARTS - 1) k = PARTS - 1;
            idx = k;
            atomicAdd(&s_cnt[k], 1.0f);                // ds_add_f32
            atomicAdd(&s_sd[k], dist);
            atomicAdd(&s_sr[k], dir);
        }
        s_idx[n] = idx;
    }
    __syncthreads();

    if (tid < PARTS) s_inv[tid] = 1.0f / (s_cnt[tid] + 1e-4f);
    __syncthreads();

    // ---------------- Phase 2: f_scan = relu(scan_circle @ W + b) -------------
    for (int i = tid; i < PARTS * DRES; i += 128) {    // 4 iters, full EXEC
        const int p = i >> 6, d = i & (DRES - 1);
        const float inv = s_inv[p];
        float v = s_sd[p] * inv * Wce[d] + s_sr[p] * inv * Wce[DRES + d] + bce[d];
        out[(((size_t)b * PARTS + p) * 2 * DRES) + DRES + d] = fmaxf(v, 0.0f);
    }

    // ---------------- Phase 3: resonance_circle via WMMA ----------------------
    // C[16x16] += A[16x4] (one-hot^T, synthesized) x B[4x16] (f_resonance slice)
    // K-loop over 128 neighbors in steps of 4. Wave w owns dims [16w, 16w+16).
    const int dbase = wave * 16;
    const int hlf   = lane >> 4;      // 0: lanes 0-15, 1: lanes 16-31
    const int m     = lane & 15;      // A row (partition) / B,C column (dim)
    const float* fb = fres + (size_t)b * NNEI * DRES + dbase + m;

    v8f c = {};
    for (int kk = 0; kk < NNEI; kk += 4) {
        const int k0 = kk + 2 * hlf;  // this lane-half's K for VGPR0
        const int k1 = k0 + 1;        // ... for VGPR1
        v2f a, bm;
        a.x  = (s_idx[k0] == m) ? 1.0f : 0.0f;   // rows 8..15 stay zero (idx<=7)
        a.y  = (s_idx[k1] == m) ? 1.0f : 0.0f;
        bm.x = fb[(size_t)k0 * DRES];            // 64B contiguous per lane-half
        bm.y = fb[(size_t)k1 * DRES];
        // v_wmma_f32_16x16x4_f32: (neg_a, A, neg_b, B, c_mod, C, reuse_a, reuse_b)
        c = __builtin_amdgcn_wmma_f32_16x16x4_f32(false, a, false, bm,
                                                  (short)0, c, false, false);
    }

    // C layout: VGPR v, lanes 0-15 -> row M=v (partitions 0..7), col N=lane.
    if (hlf == 0) {
        #pragma unroll
        for (int p = 0; p < PARTS; ++p) {
            out[(((size_t)b * PARTS + p) * 2 * DRES) + dbase + m] = c[p] * s_inv[p];
        }
    }
}

extern "C" void kernel_launch(void* const* d_in, const int* in_sizes, int n_in,
                              void* d_out, int out_size, void* d_ws, size_t ws_size,
                              hipStream_t stream) {
    (void)n_in; (void)d_ws; (void)ws_size; (void)out_size;
    const float* ego  = (const float*)d_in[0];   // [B, T, 2]
    const float* nei  = (const float*)d_in[1];   // [B, N, T, 2]
    const float* fres = (const float*)d_in[2];   // [B, N, D]
    const float* Wce  = (const float*)d_in[3];   // [2, D]
    const float* bce  = (const float*)d_in[4];   // [D]
    float* out = (float*)d_out;                  // [B, P, 2D]

    const int B = in_sizes[0] / (TLEN * 2);      // 4096
    circle_layer_kernel<<<dim3(B), dim3(128), 0, stream>>>(ego, nei, fres, Wce, bce, out);
}